// HeteAggregator_52656299049416
// MI455X (gfx1250) — compile-verified
//
#include <hip/hip_runtime.h>

typedef float v2f  __attribute__((ext_vector_type(2)));
typedef float v8f  __attribute__((ext_vector_type(8)));
typedef unsigned int u32x4 __attribute__((ext_vector_type(4)));
typedef int   i32x4 __attribute__((ext_vector_type(4)));
typedef int   i32x8 __attribute__((ext_vector_type(8)));

#define DDIM 128
#define NNB 32
#define XS_LD 132          // padded LDS row stride (floats): TDM pads 4 DWORDs per 128
#define LRELU_SLOPE 0.2f
#define BPB 8              // b-rows per block

// Issue one TDM descriptor: DMA a 32x128 fp32 tile (row-major, stride 128) from
// global memory into LDS with a 4-DWORD pad after every 128 DWORDs -> 132-float
// LDS row stride. Tracked by TENSORcnt on the issuing wave.
__device__ __forceinline__ void tdm_load_x_tile(const float* gsrc, float* lds_dst) {
  const unsigned long long ga = (unsigned long long)(size_t)gsrc;
  const unsigned int       la = (unsigned int)(size_t)(void*)lds_dst; // LDS byte offset

  u32x4 g0;
  g0[0] = 1u;                                         // count=1, user descriptor
  g0[1] = la;                                         // lds_addr
  g0[2] = (unsigned int)(ga & 0xFFFFFFFFull);         // global_addr[31:0]
  g0[3] = (unsigned int)((ga >> 32) & 0x01FFFFFFull)  // global_addr[56:32]
        | (2u << 30);                                 // type = 2 ("image")

  i32x8 g1;
  g1[0] = (2 << 16)        // data_size = 4 bytes
        | (1 << 20)        // pad_enable
        | (6 << 22)        // pad_interval: 2^(6+1) = 128 DWORDs
        | (3 << 25);       // pad_amount: 3+1 = 4 DWORDs
  g1[1] = (DDIM << 16);    // tensor_dim0[15:0] = 128 (bits 63:48)
  g1[2] = (NNB  << 16);    // tensor_dim1[15:0] = 32  (bits 95:80)
  g1[3] = (DDIM << 16);    // tile_dim0 = 128 (bits 127:112)
  g1[4] = NNB;             // tile_dim1 = 32  (bits 143:128), tile_dim2 = 0
  g1[5] = DDIM;            // tensor_dim0_stride[31:0] = 128
  g1[6] = 0;               // stride hi / tensor_dim1_stride lo
  g1[7] = 0;

  i32x4 gz = {0, 0, 0, 0}; // groups 2/3: 2-D tile, higher dims unused

#if defined(__clang_major__) && __clang_major__ >= 23
  i32x8 gz8 = {0, 0, 0, 0, 0, 0, 0, 0};
  __builtin_amdgcn_tensor_load_to_lds(g0, g1, gz, gz, gz8, 0);
#else
  __builtin_amdgcn_tensor_load_to_lds(g0, g1, gz, gz, 0);
#endif
}

__global__ __launch_bounds__(512) void hete_agg_wmma_f32(
    const float* __restrict__ self_v,   // [B, D]
    const float* __restrict__ nbr,      // [B, N, D]
    const float* __restrict__ w,        // [D, D]  (k-major: w[k][e])
    const float* __restrict__ q,        // [D]
    float* __restrict__ out,            // [B, D]
    int B)
{
  __shared__ float Ws[DDIM * DDIM];       // 64 KB, [k][e]
  __shared__ float Xs[2][NNB * XS_LD];    // 2 x ~16.9 KB double buffer (TDM-filled)
  __shared__ float sAlpha[NNB];           // scores, then softmax weights
  __shared__ float sOut[DDIM];

  const int tid  = threadIdx.x;
  const int lane = tid & 31;
  const int wave = tid >> 5;              // 0..15
  const int mt   = wave >> 3;             // neighbor-tile 0..1
  const int nt   = wave & 7;              // feature-tile 0..7
  const int half = lane >> 4;             // 0 or 1 (upper half-wave)
  const int l16  = lane & 15;

  const int b0 = blockIdx.x * BPB;

  // ---- prefetch X_{b0} via TDM (wave 0 only; TENSORcnt tracks it) ----
  if (b0 < B && wave == 0)
    tdm_load_x_tile(nbr + (size_t)b0 * NNB * DDIM, &Xs[0][0]);

  // ---- stage W into LDS once per block (L2-resident, 64KB) ----
  {
    const float4* wg = reinterpret_cast<const float4*>(w);
    float4*       wl = reinterpret_cast<float4*>(Ws);
    for (int i = tid; i < (DDIM * DDIM) / 4; i += 512) wl[i] = wg[i];
  }
  __syncthreads();

  int buf = 0;
  for (int bi = 0; bi < BPB; ++bi) {
    const int b = b0 + bi;
    if (b >= B) break;                    // uniform across the block

    // current buffer's DMA must have landed before anyone reads it
    if (wave == 0) __builtin_amdgcn_s_wait_tensorcnt(0);
    if (tid < NNB)  sAlpha[tid] = 0.0f;
    if (tid < DDIM) sOut[tid]   = 0.0f;
    __syncthreads();                      // publishes Xs[buf] + zeroed accumulators

    // prefetch next tile into the other buffer; overlaps with the WMMA work below
    const int bn = b + 1;
    if (bn < b0 + BPB && bn < B && wave == 0)
      tdm_load_x_tile(nbr + (size_t)bn * NNB * DDIM, &Xs[buf ^ 1][0]);

    // ---- GEMM tile: acc(16x16) = X[mt] (16x128) @ W[:,nt] (128x16) ----
    // 32 x V_WMMA_F32_16X16X4_F32, K=4 per step.
    // A layout: lane -> M=l16; VGPR v holds K = k0 + v + 2*half
    // B layout: lane -> N=l16; VGPR v holds K row (k0 + v + 2*half)
    v8f acc = {};
    const float* __restrict__ Xb = &Xs[buf][0];
    const int arow = (mt * 16 + l16) * XS_LD;
    const int bcol = nt * 16 + l16;
    for (int k0 = 0; k0 < DDIM; k0 += 4) {
      const int ka = k0 + 2 * half;
      v2f a, bf;
      a.x  = Xb[arow + ka];
      a.y  = Xb[arow + ka + 1];
      bf.x = Ws[ka * DDIM + bcol];
      bf.y = Ws[(ka + 1) * DDIM + bcol];
      acc = __builtin_amdgcn_wmma_f32_16x16x4_f32(
          /*neg_a=*/false, a, /*neg_b=*/false, bf,
          /*c_mod=*/(short)0, acc, /*reuse_a=*/false, /*reuse_b=*/false);
    }

    // ---- attention scores: score[n] = sum_d q[d] * leaky(self[b,d] * m[n,d]) ----
    // C layout: lane covers column d = nt*16+l16; row M = r + 8*half (+ mt*16)
    const int   d  = nt * 16 + l16;
    const float sv = self_v[(size_t)b * DDIM + d];
    const float qd = q[d];
#pragma unroll
    for (int r = 0; r < 8; ++r) {
      float v = acc[r] * sv;
      v = (v >= 0.0f) ? v : (LRELU_SLOPE * v);
      v *= qd;
      // reduce the 16 lanes that share this neighbor row
      v += __shfl_xor(v, 8, 32);
      v += __shfl_xor(v, 4, 32);
      v += __shfl_xor(v, 2, 32);
      v += __shfl_xor(v, 1, 32);
      if (l16 == 0) atomicAdd(&sAlpha[mt * 16 + r + 8 * half], v);
    }
    __syncthreads();

    // ---- softmax over the 32 neighbors (wave 0, wave32 shuffles) ----
    if (wave == 0) {
      float s = sAlpha[lane];
      float m = s;
      m = fmaxf(m, __shfl_xor(m, 16, 32));
      m = fmaxf(m, __shfl_xor(m, 8, 32));
      m = fmaxf(m, __shfl_xor(m, 4, 32));
      m = fmaxf(m, __shfl_xor(m, 2, 32));
      m = fmaxf(m, __shfl_xor(m, 1, 32));
      float e = __expf(s - m);
      float sum = e;
      sum += __shfl_xor(sum, 16, 32);
      sum += __shfl_xor(sum, 8, 32);
      sum += __shfl_xor(sum, 4, 32);
      sum += __shfl_xor(sum, 2, 32);
      sum += __shfl_xor(sum, 1, 32);
      sAlpha[lane] = e / sum;
    }
    __syncthreads();

    // ---- weighted sum over neighbors: out[d] = sum_n alpha[n] * m[n,d] ----
    float po = 0.0f;
#pragma unroll
    for (int r = 0; r < 8; ++r)
      po += sAlpha[mt * 16 + r + 8 * half] * acc[r];
    po += __shfl_xor(po, 16, 32);         // lanes L and L+16 share the same d
    if (half == 0) atomicAdd(&sOut[d], po);
    __syncthreads();

    if (tid < DDIM) out[(size_t)b * DDIM + tid] = sOut[tid];
    __syncthreads();                      // protect LDS reuse next iteration

    buf ^= 1;
  }
}

extern "C" void kernel_launch(void* const* d_in, const int* in_sizes, int n_in,
                              void* d_out, int out_size, void* d_ws, size_t ws_size,
                              hipStream_t stream) {
  const float* self_v = (const float*)d_in[0];   // [B, 128]
  const float* nbr    = (const float*)d_in[1];   // [B, 32, 128]
  const float* w      = (const float*)d_in[2];   // [128, 128]
  const float* q      = (const float*)d_in[3];   // [128, 1]
  float* out          = (float*)d_out;           // [B, 128]

  const int B = in_sizes[0] / DDIM;              // 20000
  const int grid = (B + BPB - 1) / BPB;          // 2500 blocks
  hete_agg_wmma_f32<<<grid, 512, 0, stream>>>(self_v, nbr, w, q, out, B);
}